// DummyControlModel_3831110828037
// MI455X (gfx1250) — compile-verified
//
#include <hip/hip_runtime.h>

typedef __attribute__((ext_vector_type(2))) float v2f;
typedef __attribute__((ext_vector_type(8))) float v8f;

namespace {

constexpr float kDT = 1.0f / 10.0f;  // TIME_DURATION / NUM_SEGMENT

struct Scalars {
  float r, inv_r, ReH01sq;
  int big;  // rsq > 1e-24
};

// Derive the Hamiltonian scalars from the 4-element para_ham_unknown.
// herm = 0.5(x+x^T) + 0.5i(x-x^T); H = 0.5*sigma_x + (herm - tr(herm)/2 * I);
// H0 = H - tr(H)/2 * I; rsq = tr(H0 H0).real / 2;  g = H0_01 + H0_10 = 2*Re(H0_01).
__device__ inline Scalars make_scalars(const float* __restrict__ para) {
  float x00 = para[0], x01 = para[1], x10 = para[2], x11 = para[3];
  float trh = x00 + x11;
  float H00 = x00 - 0.5f * trh;
  float H11 = x11 - 0.5f * trh;
  float Re01 = 0.5f + 0.5f * (x01 + x10);   // includes HAM_MODEL's 0.5
  float Im01 = 0.5f * (x01 - x10);
  float tr = H00 + H11;                      // ~0, kept for generality
  float h00 = H00 - 0.5f * tr;
  float h11 = H11 - 0.5f * tr;
  float rsq = 0.5f * (h00 * h00 + h11 * h11 + 2.0f * (Re01 * Re01 + Im01 * Im01));
  Scalars s;
  s.r = sqrtf(fmaxf(rsq, 1e-30f));
  s.inv_r = 1.0f / s.r;
  s.ReH01sq = Re01 * Re01;                   // g^2/4 with g = 2*Re(H0_01)
  s.big = rsq > 1e-24f;
  return s;
}

// Commuting exponentials collapse: U = exp(-i*Phi*H), Phi = DT * sum(omega_row).
// infidelity = 1 - |u01+u10|^2/4 = 1 - (sin(Phi*r)/r)^2 * Re(H0_01)^2  (|phase|=1).
__device__ inline float row_infid(float S, const Scalars& sc) {
  float phi = kDT * S;
  float theta = phi * sc.r;
  float sinc = sc.big ? __sinf(theta) * sc.inv_r : phi;
  return 1.0f - sinc * sinc * sc.ReH01sq;
}

__device__ inline float block_reduce(float acc, int lane, int wv, int nwaves) {
  #pragma unroll
  for (int off = 16; off > 0; off >>= 1) acc += __shfl_xor(acc, off, 32);
  __shared__ float smem[8];
  if (lane == 0) smem[wv] = acc;
  __syncthreads();
  float s = 0.0f;
  if (threadIdx.x == 0) {
    for (int i = 0; i < nwaves; ++i) s += smem[i];
  }
  return s;
}

}  // namespace

// Main streaming kernel: each wave processes two 16-row tiles per iteration.
// Row segment sums are produced by V_WMMA_F32_16X16X4_F32 with an all-ones A
// matrix (D[m][n] = column-sum of B, so every lane reads its row's sum from c[0]).
__global__ __launch_bounds__(256) void qc_main(
    const float* __restrict__ para, const float* __restrict__ omega,
    const float* __restrict__ infd, float* __restrict__ partial, int ntiles) {
  const Scalars sc = make_scalars(para);
  const int lane = threadIdx.x & 31;
  const int wv = threadIdx.x >> 5;
  const int waves_per_blk = blockDim.x >> 5;
  const int gw = blockIdx.x * waves_per_blk + wv;
  const int nw = gridDim.x * waves_per_blk;
  const int col = lane & 15;
  const bool lo = lane < 16;
  const v2f ones = {1.0f, 1.0f};
  const v2f zero2 = {0.0f, 0.0f};

  float acc = 0.0f;
  for (int t = gw; t < ntiles; t += nw) {
    const long long base = (long long)t * 32;
    // Tile 0 = rows [base, base+16), tile 1 = rows [base+16, base+32).
    // Lane L supplies B column (L&15); low lanes K-slots {c,c+1}, high {c+2,c+3}.
    const float* p0 = omega + (base + col) * 10 + (lo ? 0 : 2);
    const float* p1 = p0 + 160;  // +16 rows * 10 segments

    v2f b00 = __builtin_nontemporal_load((const v2f*)(p0));
    v2f b01 = __builtin_nontemporal_load((const v2f*)(p0 + 4));
    v2f b02 = lo ? __builtin_nontemporal_load((const v2f*)(p0 + 8)) : zero2;  // segs 8,9; pad 0
    v2f b10 = __builtin_nontemporal_load((const v2f*)(p1));
    v2f b11 = __builtin_nontemporal_load((const v2f*)(p1 + 4));
    v2f b12 = lo ? __builtin_nontemporal_load((const v2f*)(p1 + 8)) : zero2;

    v8f c0 = {};
    v8f c1 = {};
    c0 = __builtin_amdgcn_wmma_f32_16x16x4_f32(false, ones, false, b00, (short)0, c0, false, false);
    c0 = __builtin_amdgcn_wmma_f32_16x16x4_f32(false, ones, false, b01, (short)0, c0, false, false);
    c0 = __builtin_amdgcn_wmma_f32_16x16x4_f32(false, ones, false, b02, (short)0, c0, false, false);
    c1 = __builtin_amdgcn_wmma_f32_16x16x4_f32(false, ones, false, b10, (short)0, c1, false, false);
    c1 = __builtin_amdgcn_wmma_f32_16x16x4_f32(false, ones, false, b11, (short)0, c1, false, false);
    c1 = __builtin_amdgcn_wmma_f32_16x16x4_f32(false, ones, false, b12, (short)0, c1, false, false);

    // c[0]: lanes 0-15 hold D[0][lane] (tile0 colsums); lanes 16-31 hold
    // D[8][lane-16] (tile1 colsums). Either way: row (base+lane)'s segment sum.
    float S = lo ? c0[0] : c1[0];
    float infid = row_infid(S, sc);
    float id = __builtin_nontemporal_load(infd + base + lane);
    float d = id - infid;
    acc = fmaf(d, d, acc);
  }

  float s = block_reduce(acc, lane, wv, waves_per_blk);
  if (threadIdx.x == 0) partial[blockIdx.x] = s;
}

// Single-block finisher: deterministic sum of per-block partials + scalar tail.
__global__ __launch_bounds__(256) void qc_finish(
    const float* __restrict__ para, const float* __restrict__ omega,
    const float* __restrict__ infd, const float* __restrict__ partial,
    int nparts, long long tail_start, long long n, float* __restrict__ out) {
  const Scalars sc = make_scalars(para);
  float acc = 0.0f;
  for (int i = threadIdx.x; i < nparts; i += blockDim.x) acc += partial[i];
  for (long long b = tail_start + threadIdx.x; b < n; b += blockDim.x) {
    const float* p = omega + b * 10;
    float S = 0.0f;
    #pragma unroll
    for (int s = 0; s < 10; ++s) S += p[s];
    float d = infd[b] - row_infid(S, sc);
    acc = fmaf(d, d, acc);
  }
  const int lane = threadIdx.x & 31;
  const int wv = threadIdx.x >> 5;
  float s = block_reduce(acc, lane, wv, blockDim.x >> 5);
  if (threadIdx.x == 0) out[0] = s / (float)n;
}

extern "C" void kernel_launch(void* const* d_in, const int* in_sizes, int n_in,
                              void* d_out, int out_size, void* d_ws, size_t ws_size,
                              hipStream_t stream) {
  const float* para = (const float*)d_in[0];   // (2,2) float32
  const float* omega = (const float*)d_in[1];  // (n,10) float32
  const float* infd = (const float*)d_in[2];   // (n,)   float32
  float* out = (float*)d_out;

  const long long n = (long long)in_sizes[2];
  const long long ntiles = n / 32;             // full 32-row tiles (n=2e6 -> exact)
  float* partial = (float*)d_ws;

  int blocks = 2048;                            // 8 waves/block -> 16K waves
  const long long max_ws_blocks = (long long)(ws_size / sizeof(float));
  if ((long long)blocks > max_ws_blocks) blocks = (int)(max_ws_blocks > 0 ? max_ws_blocks : 1);
  const long long waves_needed = ntiles > 0 ? ((ntiles + 7) / 8) : 1;
  if ((long long)blocks > waves_needed) blocks = (int)waves_needed;
  if (blocks < 1) blocks = 1;

  qc_main<<<blocks, 256, 0, stream>>>(para, omega, infd, partial, (int)ntiles);
  qc_finish<<<1, 256, 0, stream>>>(para, omega, infd, partial, blocks,
                                   ntiles * 32, n, out);
}